// GenericTreeLSTMCell_57578331570339
// MI455X (gfx1250) — compile-verified
//
#include <hip/hip_runtime.h>

typedef float v2f __attribute__((ext_vector_type(2)));
typedef float v8f __attribute__((ext_vector_type(8)));

#define NNODES     50000
#define HDIM       256
#define KTOT       768              // 3 real children * 256
#define NSUB       4                // 16-node subtiles per workgroup
#define LDS_STRIDE 68               // 64 + 4 pad -> conflict-free b64 frag reads
#define IOU_STEPS  192              // KTOT / 4
#define F_STEPS    64               // 256 / 4
#define IOU_ITEMS  (16 * 3 * IOU_STEPS * 32)   // float2 items in iou pack
#define F_ITEMS    (16 * 3 * F_STEPS * 32)     // float2 items in f pack

static __device__ __forceinline__ v8f wmma_f32(v2f a, v2f b, v8f c) {
  // D = A(16x4) * B(4x16) + C(16x16), fp32 matrix core path on gfx1250
  return __builtin_amdgcn_wmma_f32_16x16x4_f32(false, a, false, b, (short)0, c,
                                               false, false);
}

static __device__ __forceinline__ float sigmoidf(float x) {
  return 1.0f / (1.0f + __expf(-x));
}

// ---------------------------------------------------------------------------
// Repack U_iou [4,768,256] and U_f [4,256,256] (torch [out,in] convention)
// into per-lane WMMA B-fragment order:
//   B row K striped: VGPR0 = {K0: lanes0-15, K2: lanes16-31}, VGPR1 = {K1,K3}
// ---------------------------------------------------------------------------
__global__ void pack_weights(const float* __restrict__ U_f,
                             const float* __restrict__ U_iou,
                             float* __restrict__ wpack) {
  const int id = blockIdx.x * blockDim.x + threadIdx.x;
  if (id >= IOU_ITEMS + F_ITEMS) return;
  if (id < IOU_ITEMS) {
    const int lane = id & 31;
    const int s    = (id >> 5) % IOU_STEPS;
    const int g    = ((id >> 5) / IOU_STEPS) % 3;       // 0=i, 1=o, 2=u
    const int kt   = id / (32 * IOU_STEPS * 3);
    const int col  = g * 256 + kt * 16 + (lane & 15);   // output column in [0,768)
    const int jj   = s * 4 + 2 * (lane >> 4);           // global K index
    const int d    = jj >> 8;                           // child
    const int h    = jj & 255;
    const size_t src = ((size_t)d * 768 + col) * 256 + h;
    wpack[2 * (size_t)id]     = U_iou[src];
    wpack[2 * (size_t)id + 1] = U_iou[src + 1];
  } else {
    const int id2  = id - IOU_ITEMS;
    const int lane = id2 & 31;
    const int s    = (id2 >> 5) % F_STEPS;
    const int d    = ((id2 >> 5) / F_STEPS) % 3;
    const int kt   = id2 / (32 * F_STEPS * 3);
    const int col  = kt * 16 + (lane & 15);
    const int h    = s * 4 + 2 * (lane >> 4);
    const size_t src = ((size_t)d * 256 + col) * 256 + h;
    wpack[2 * (size_t)id]     = U_f[src];
    wpack[2 * (size_t)id + 1] = U_f[src + 1];
  }
}

// ---------------------------------------------------------------------------
// Async A-chunk stage: 64 nodes x 64 K into LDS via the CDNA5 async
// global->LDS DMA path (ASYNCcnt, no VGPR round-trip). Each lane moves 16B.
// ---------------------------------------------------------------------------
static __device__ __forceinline__ void load_chunk_async(float* dst,
                                                        const float* __restrict__ nh,
                                                        int nodebase, int chunk,
                                                        int tid) {
  const int j0 = chunk * 64;
#pragma unroll
  for (int it = 0; it < 2; ++it) {
    const int r = (tid >> 4) + it * 32;   // node row 0..63
    const int c = (tid & 15) * 4;         // K offset 0..60
    int node = nodebase + r;
    if (node >= NNODES) node = NNODES - 1;   // clamp; garbage rows never stored
    const unsigned long long gaddr =
        (unsigned long long)(uintptr_t)(nh + (size_t)node * KTOT + j0 + c);
    const unsigned ldsoff =
        (unsigned)(uintptr_t)(dst + r * LDS_STRIDE + c);  // LDS byte offset
    asm volatile("global_load_async_to_lds_b128 %0, %1, off"
                 :
                 : "v"(ldsoff), "v"(gaddr)
                 : "memory");
  }
}

static __device__ __forceinline__ void wait_async0() {
  asm volatile("s_wait_asynccnt 0x0" ::: "memory");
}

// B fragments for one K-step (i/o/u from U_iou pack, f from U_f pack).
struct Bfrags { v2f i, o, u, f; };

static __device__ __forceinline__ Bfrags load_b(const float* __restrict__ wpack,
                                                int kt, int lane, int gstep) {
  Bfrags r;
  const size_t ib = 2 * ((size_t)((kt * 3 + 0) * IOU_STEPS + gstep) * 32 + lane);
  r.i = *reinterpret_cast<const v2f*>(wpack + ib);
  r.o = *reinterpret_cast<const v2f*>(wpack + ib + 2 * (size_t)IOU_STEPS * 32);
  r.u = *reinterpret_cast<const v2f*>(wpack + ib + 4 * (size_t)IOU_STEPS * 32);
  const int d  = gstep >> 6;
  const int fs = gstep & 63;
  const size_t fb =
      2 * ((size_t)IOU_ITEMS + (size_t)((kt * 3 + d) * F_STEPS + fs) * 32 + lane);
  r.f = *reinterpret_cast<const v2f*>(wpack + fb);
  return r;
}

// ---------------------------------------------------------------------------
// Main fused tree-LSTM kernel. Block = 512 threads (16 wave32).
// Wave w owns column tile kt=w (16 cols); per M-subtile it accumulates
// i/o/u tiles over K=768 and the per-child f tile over K=256, folding f
// into c_aggr at each child boundary, then does the full gate epilogue.
// B fragments are software-pipelined one K-step ahead to hide L2 latency.
// ---------------------------------------------------------------------------
__global__ __launch_bounds__(512, 1)
void treelstm_wmma(const float* __restrict__ nh, const float* __restrict__ nc,
                   const float* __restrict__ f_in, const float* __restrict__ iou_in,
                   const float* __restrict__ wpack, float* __restrict__ out) {
  __shared__ float lds_a[2][64 * LDS_STRIDE];

  const int tid  = threadIdx.x;
  const int lane = tid & 31;
  const int kt   = tid >> 5;          // wave id == column tile
  const int rowL = lane & 15;
  const int kgrp = lane >> 4;
  const int nodebase = blockIdx.x * (NSUB * 16);
  const int col = kt * 16 + rowL;     // this lane's output column (C-tile N)

  v8f acc_i[NSUB], acc_o[NSUB], acc_u[NSUB], acc_f[NSUB], c_aggr[NSUB];
#pragma unroll
  for (int m = 0; m < NSUB; ++m)
#pragma unroll
    for (int v = 0; v < 8; ++v) {
      acc_i[m][v] = 0.f; acc_o[m][v] = 0.f; acc_u[m][v] = 0.f;
      acc_f[m][v] = 0.f; c_aggr[m][v] = 0.f;
    }

  load_chunk_async(lds_a[0], nh, nodebase, 0, tid);
  Bfrags bcur = load_b(wpack, kt, lane, 0);   // prime B pipeline
  wait_async0();
  __syncthreads();

  for (int chunk = 0; chunk < 12; ++chunk) {
    const int buf = chunk & 1;
    if (chunk + 1 < 12)
      load_chunk_async(lds_a[buf ^ 1], nh, nodebase, chunk + 1, tid);
    const int d = chunk >> 2;  // current child

    for (int s = 0; s < 16; ++s) {
      const int gstep = chunk * 16 + s;             // 0..191
      // prefetch next step's B fragments (clamped; redundant last load is ok)
      const int gnext = (gstep + 1 < IOU_STEPS) ? gstep + 1 : IOU_STEPS - 1;
      const Bfrags bnext = load_b(wpack, kt, lane, gnext);

#pragma unroll
      for (int m = 0; m < NSUB; ++m) {
        const int word = (m * 16 + rowL) * LDS_STRIDE + s * 4 + 2 * kgrp;
        const v2f a = *reinterpret_cast<const v2f*>(&lds_a[buf][word]);
        acc_i[m] = wmma_f32(a, bcur.i, acc_i[m]);
        acc_o[m] = wmma_f32(a, bcur.o, acc_o[m]);
        acc_u[m] = wmma_f32(a, bcur.u, acc_u[m]);
        acc_f[m] = wmma_f32(a, bcur.f, acc_f[m]);
      }
      bcur = bnext;
    }

    // Child boundary: fold forget gate into c_aggr, free f accumulators.
    if ((chunk & 3) == 3) {
#pragma unroll
      for (int m = 0; m < NSUB; ++m)
#pragma unroll
        for (int v = 0; v < 8; ++v) {
          int node = nodebase + m * 16 + v + 8 * kgrp;   // C-tile row M
          if (node >= NNODES) node = NNODES - 1;          // clamp (never stored)
          const float flog = acc_f[m][v] + f_in[(size_t)node * HDIM + col];
          const float fg   = sigmoidf(flog);
          c_aggr[m][v] += fg * nc[((size_t)node * 3 + d) * HDIM + col];
          acc_f[m][v] = 0.f;
        }
    }
    wait_async0();        // chunk+1 A data resident in LDS
    __syncthreads();
  }

  // Epilogue: iou counted twice per reference; c = sig(i)*tanh(u)+c_aggr,
  // h = sig(o)*tanh(c). Output = [h (N*H) | c (N*H)].
#pragma unroll
  for (int m = 0; m < NSUB; ++m)
#pragma unroll
    for (int v = 0; v < 8; ++v) {
      const int node = nodebase + m * 16 + v + 8 * kgrp;
      if (node < NNODES) {
        const size_t b = (size_t)node * KTOT;
        const float iv = acc_i[m][v] + 2.f * iou_in[b + col];
        const float ov = acc_o[m][v] + 2.f * iou_in[b + 256 + col];
        const float uv = acc_u[m][v] + 2.f * iou_in[b + 512 + col];
        const float c  = sigmoidf(iv) * tanhf(uv) + c_aggr[m][v];
        const float h  = sigmoidf(ov) * tanhf(c);
        out[(size_t)node * HDIM + col] = h;
        out[(size_t)NNODES * HDIM + (size_t)node * HDIM + col] = c;
      }
    }
}

extern "C" void kernel_launch(void* const* d_in, const int* in_sizes, int n_in,
                              void* d_out, int out_size, void* d_ws, size_t ws_size,
                              hipStream_t stream) {
  (void)in_sizes; (void)n_in; (void)out_size; (void)ws_size;
  const float* nh     = (const float*)d_in[0];  // [N,3,256]
  const float* nc     = (const float*)d_in[1];  // [N,3,256]
  const float* f_in   = (const float*)d_in[2];  // [N,256]
  const float* iou_in = (const float*)d_in[3];  // [N,768]
  const float* U_f    = (const float*)d_in[4];  // [4,256,256]
  const float* U_iou  = (const float*)d_in[5];  // [4,768,256]
  float* out   = (float*)d_out;                 // [2*N*256]
  float* wpack = (float*)d_ws;                  // 786432 floats = 3 MB

  const int packItems = IOU_ITEMS + F_ITEMS;    // 393216
  pack_weights<<<(packItems + 255) / 256, 256, 0, stream>>>(U_f, U_iou, wpack);
  treelstm_wmma<<<(NNODES + NSUB * 16 - 1) / (NSUB * 16), 512, 0, stream>>>(
      nh, nc, f_in, iou_in, wpack, out);
}